// Attn_19524921327936
// MI455X (gfx1250) — compile-verified
//
#include <hip/hip_runtime.h>
#include <hip/hip_bf16.h>
#include <math.h>

// Problem constants (match reference): N=32 batches, S=8192 seq, H=512 hidden
#define NB 32
#define SS 8192
#define HH 512

typedef float v2f __attribute__((ext_vector_type(2)));
typedef float v8f __attribute__((ext_vector_type(8)));

// ---------------------------------------------------------------------------
// Kernel 1: batched GEMV via V_WMMA_F32_16X16X4_F32.
//   grid = (S/128, N), block = 256 threads = 8 wave32s.
//   Wave w computes energies for 16 seq rows: s0 = bx*128 + w*16.
//   A tile:  enc[n, s0+M, hb..hb+3]   (16x4 f32, striped per ISA layout)
//   B tile:  hidden[n, hb..hb+3] replicated across all 16 columns
//   -> D[M, j] identical across j; columns give the 16 dot-product partials.
// Writes energies straight into d_out with the final [s*NB + n] layout.
// ---------------------------------------------------------------------------
__global__ void __launch_bounds__(256)
attn_energies_wmma(const float* __restrict__ hidden,
                   const float* __restrict__ enc,
                   float* __restrict__ out /* [S, N] energies */) {
    __shared__ float sh_hidden[HH];

    const int n    = blockIdx.y;
    const int tid  = threadIdx.x;
    const int wave = tid >> 5;        // 0..7
    const int lane = tid & 31;        // wave32
    const int half = lane >> 4;       // 0: K=0..1, 1: K=2..3
    const int row  = lane & 15;       // M within 16-row tile

    // Stage hidden[n] (2 KB) into LDS once; reused by 128 WMMA steps x 8 waves.
    for (int i = tid; i < HH; i += 256)
        sh_hidden[i] = hidden[(size_t)n * HH + i];
    __syncthreads();

    const int s0 = blockIdx.x * 128 + wave * 16;
    const float* arow = enc + ((size_t)n * SS + (size_t)(s0 + row)) * HH
                            + (size_t)(half * 2);

    v8f c = {0.f, 0.f, 0.f, 0.f, 0.f, 0.f, 0.f, 0.f};

#pragma unroll 8
    for (int kb = 0; kb < HH / 4; ++kb) {
        const int hb = kb * 4;
        // A: lane's 2 consecutive f32 of its seq row (K = half*2, half*2+1)
        v2f a = *(const v2f*)(arow + hb);
        // B: same 2 f32 of hidden[n] for every column (broadcast from LDS)
        v2f b = *(const v2f*)(sh_hidden + hb + half * 2);
        // D = A x B + C  (8 args: neg_a, A, neg_b, B, c_mod, C, reuse_a, reuse_b)
        c = __builtin_amdgcn_wmma_f32_16x16x4_f32(
                false, a, false, b, (short)0, c, false, false);
    }

    // C/D layout: lanes 0-15 hold M=0..7 in c[0..7] (N=lane), lanes 16-31 hold
    // M=8..15. All columns identical, so lane 0 and lane 16 emit the 16 rows.
    if ((lane & 15) == 0) {
#pragma unroll
        for (int m = 0; m < 8; ++m) {
            const int s = s0 + half * 8 + m;
            out[(size_t)s * NB + n] = c[m];
        }
    }
}

// ---------------------------------------------------------------------------
// Kernel 2: softmax over the S axis, per batch column n. In-place on d_out.
//   One block per n; 256 threads x 32 values each. Values cached in registers,
//   so reads fully precede the post-reduction writes (in-place safe).
// ---------------------------------------------------------------------------
__global__ void __launch_bounds__(256)
softmax_over_s(float* __restrict__ out /* [S, N] in, [S, N] out */) {
    __shared__ float red[256];

    const int n = blockIdx.x;
    const int t = threadIdx.x;
    constexpr int PER = SS / 256;     // 32 values per thread

    float v[PER];
    float m = -3.402823466e38f;
#pragma unroll
    for (int i = 0; i < PER; ++i) {
        v[i] = out[(size_t)(t + i * 256) * NB + n];
        m = fmaxf(m, v[i]);
    }

    // max reduction
    red[t] = m;
    __syncthreads();
    for (int off = 128; off > 0; off >>= 1) {
        if (t < off) red[t] = fmaxf(red[t], red[t + off]);
        __syncthreads();
    }
    const float gmax = red[0];
    __syncthreads();

    float s = 0.f;
#pragma unroll
    for (int i = 0; i < PER; ++i) {
        v[i] = __expf(v[i] - gmax);
        s += v[i];
    }

    // sum reduction
    red[t] = s;
    __syncthreads();
    for (int off = 128; off > 0; off >>= 1) {
        if (t < off) red[t] += red[t + off];
        __syncthreads();
    }
    const float inv = 1.0f / red[0];
    __syncthreads();

#pragma unroll
    for (int i = 0; i < PER; ++i)
        out[(size_t)(t + i * 256) * NB + n] = v[i] * inv;
}

extern "C" void kernel_launch(void* const* d_in, const int* in_sizes, int n_in,
                              void* d_out, int out_size, void* d_ws, size_t ws_size,
                              hipStream_t stream) {
    const float* hidden = (const float*)d_in[0];   // [N, H] f32
    const float* enc    = (const float*)d_in[1];   // [N, S, H] f32
    float* out          = (float*)d_out;           // [S, N, 1] f32

    dim3 grid1(SS / 128, NB);
    attn_energies_wmma<<<grid1, 256, 0, stream>>>(hidden, enc, out);
    softmax_over_s<<<NB, 256, 0, stream>>>(out);
}